// CRF_62448824484549
// MI455X (gfx1250) — compile-verified
//
#include <hip/hip_runtime.h>
#include <math.h>

typedef __attribute__((ext_vector_type(2))) float v2f;
typedef __attribute__((ext_vector_type(8))) float v8f;

#define SEQ_LEN 512
#define BATCH   1024
#define NTAGS   48
#define QPITCH  52   // padded row pitch (floats) for bank-conflict-free ds_load_b64

#define LOG2E 1.4426950408889634f
#define LN2   0.6931471805599453f

// Raw hardware transcendentals (v_exp_f32 = 2^x, v_log_f32 = log2 x).
// Safe here: GEMM output S in [~0.9, ~53] (never <=0 / denormal), exp args <= 0.
__device__ __forceinline__ float fast_exp2(float x) { return __builtin_amdgcn_exp2f(x); }
__device__ __forceinline__ float fast_log2(float x) { return __builtin_amdgcn_logf(x); }
__device__ __forceinline__ float fast_exp(float x)  { return __builtin_amdgcn_exp2f(x * LOG2E); }

// ---------------------------------------------------------------------------
// Partition (forward algorithm) kernel — one wave per workgroup, each wave
// owns 16 batch columns for all 512 steps. Recurrence is computed as
// S = expT^T (48x48) x q^T (48x16) with V_WMMA_F32_16X16X4_F32:
// 3 M-tiles x 12 K-chunks = 36 WMMAs per step. Whole recurrence tracked in
// base-2 log domain so each element costs exactly one v_log + one v_exp.
// ---------------------------------------------------------------------------
__global__ __launch_bounds__(32) void crf_partition_kernel(
    const float* __restrict__ emissions,   // [S, B, T]
    const int*   __restrict__ mask,        // [S, B]
    const float* __restrict__ trans,       // [T, T]
    const float* __restrict__ startT,      // [T]
    const float* __restrict__ endT,        // [T]
    float*       __restrict__ part_out)    // [B]
{
    __shared__ float q[16 * QPITCH];

    const int lane = threadIdx.x & 31;
    const int l16  = lane & 15;   // batch column within the wave's tile
    const int hi   = lane >> 4;   // lane half (selects K/M sub-stripe)
    const int b    = blockIdx.x * 16 + l16;   // global batch column

    // ---- A operand: expT^T in WMMA 32-bit A layout (16x4 per chunk) ----
    //   A.x: lanes 0-15 -> (M=l, K=4kk),   lanes 16-31 -> (M=l16, K=4kk+2)
    //   A.y: K = +1 of A.x
    v2f A[3][12];
#pragma unroll
    for (int mt = 0; mt < 3; ++mt) {
#pragma unroll
        for (int kk = 0; kk < 12; ++kk) {
            const int i0 = 4 * kk + 2 * hi;
            const int j  = 16 * mt + l16;
            A[mt][kk].x = fast_exp(trans[i0 * NTAGS + j]);
            A[mt][kk].y = fast_exp(trans[(i0 + 1) * NTAGS + j]);
        }
    }

    // ---- t = 0: alpha0 = start + em[0]; normalize (base-2 domain) ----
    float c2;  // running log2-normalizer for this batch column
    {
        const float* er = emissions + (size_t)b * NTAGS;
        float u[3][8];
        float mloc = -INFINITY;
#pragma unroll
        for (int mt = 0; mt < 3; ++mt) {
#pragma unroll
            for (int r = 0; r < 8; ++r) {
                const int j = 16 * mt + r + 8 * hi;
                u[mt][r] = (startT[j] + er[j]) * LOG2E;   // log2-domain alpha0
                mloc = fmaxf(mloc, u[mt][r]);
            }
        }
        const float M = fmaxf(mloc, __shfl_xor(mloc, 16, 32));
        c2 = M;
#pragma unroll
        for (int mt = 0; mt < 3; ++mt) {
#pragma unroll
            for (int r = 0; r < 8; ++r) {
                const int j = 16 * mt + r + 8 * hi;
                q[l16 * QPITCH + j] = fast_exp2(u[mt][r] - M);
            }
        }
    }

    // ---- forward recurrence t = 1 .. 511 ----
    for (int t = 1; t < SEQ_LEN; ++t) {
        // B operand: q^T chunks from LDS (ds_load_b64 pairs).
        v2f Bv[12];
#pragma unroll
        for (int kk = 0; kk < 12; ++kk) {
            Bv[kk] = *(const v2f*)&q[l16 * QPITCH + 4 * kk + 2 * hi];
        }

        // GEMM: S[mt] = sum_kk A[mt][kk] x Bv[kk]
        v8f acc[3];
#pragma unroll
        for (int mt = 0; mt < 3; ++mt) {
            v8f cc = {};
#pragma unroll
            for (int kk = 0; kk < 12; ++kk) {
                cc = __builtin_amdgcn_wmma_f32_16x16x4_f32(
                        false, A[mt][kk], false, Bv[kk],
                        (short)0, cc, false, false);
            }
            acc[mt] = cc;
        }

        const float* er = emissions + ((size_t)t * BATCH + b) * NTAGS;
        if (t + 1 < SEQ_LEN)
            __builtin_prefetch(emissions + ((size_t)(t + 1) * BATCH + b) * NTAGS, 0, 3);
        const int mk = mask[t * BATCH + b];

        // u2 = log2(S) + em*log2e ; per-column max over all 48 tags
        float u[3][8];
        float mloc = -INFINITY;
#pragma unroll
        for (int mt = 0; mt < 3; ++mt) {
#pragma unroll
            for (int r = 0; r < 8; ++r) {
                const int j = 16 * mt + r + 8 * hi;
                u[mt][r] = fmaf(er[j], LOG2E, fast_log2(acc[mt][r]));
                mloc = fmaxf(mloc, u[mt][r]);
            }
        }
        const float M = fmaxf(mloc, __shfl_xor(mloc, 16, 32));

        if (mk) c2 += M;
#pragma unroll
        for (int mt = 0; mt < 3; ++mt) {
#pragma unroll
            for (int r = 0; r < 8; ++r) {
                const int j = 16 * mt + r + 8 * hi;
                const float qn = fast_exp2(u[mt][r] - M);
                if (mk) q[l16 * QPITCH + j] = qn;  // masked cols keep old q
            }
        }
    }

    // ---- finalize: partition_b = ln2 * (c2 + log2(sum_j q * exp(end_j))) ----
    float sloc = 0.0f;
#pragma unroll
    for (int mt = 0; mt < 3; ++mt) {
#pragma unroll
        for (int r = 0; r < 8; ++r) {
            const int j = 16 * mt + r + 8 * hi;
            sloc += q[l16 * QPITCH + j] * fast_exp(endT[j]);
        }
    }
    const float stot = sloc + __shfl_xor(sloc, 16, 32);
    if (hi == 0) part_out[b] = LN2 * (c2 + fast_log2(stot));
}

// ---------------------------------------------------------------------------
// Gold-path score: one thread per batch element.
// ---------------------------------------------------------------------------
__global__ __launch_bounds__(256) void crf_score_kernel(
    const float* __restrict__ emissions,
    const int*   __restrict__ tags,
    const int*   __restrict__ mask,
    const float* __restrict__ trans,
    const float* __restrict__ startT,
    const float* __restrict__ endT,
    float*       __restrict__ score_out)
{
    const int b = blockIdx.x * blockDim.x + threadIdx.x;
    if (b >= BATCH) return;

    int tprev = tags[b];
    float sc  = startT[tprev] + emissions[(size_t)b * NTAGS + tprev];
    int cnt   = mask[b];
    for (int t = 1; t < SEQ_LEN; ++t) {
        const int tc = tags[t * BATCH + b];
        const int mi = mask[t * BATCH + b];
        cnt += mi;
        sc  += (trans[tprev * NTAGS + tc] +
                emissions[((size_t)t * BATCH + b) * NTAGS + tc]) * (float)mi;
        tprev = tc;
    }
    const int last_idx = (cnt > 0) ? (cnt - 1) : 0;
    const int last_tag = tags[(size_t)last_idx * BATCH + b];
    sc += endT[last_tag];
    score_out[b] = sc;
}

// ---------------------------------------------------------------------------
// Deterministic fixed-order reduction: out = sum_b (part[b] - score[b])
// ---------------------------------------------------------------------------
__global__ __launch_bounds__(256) void crf_reduce_kernel(
    const float* __restrict__ part,
    const float* __restrict__ score,
    float*       __restrict__ out)
{
    __shared__ float red[256];
    const int tid = threadIdx.x;
    float s = 0.0f;
    for (int i = tid; i < BATCH; i += 256) s += part[i] - score[i];
    red[tid] = s;
    __syncthreads();
    for (int off = 128; off > 0; off >>= 1) {
        if (tid < off) red[tid] += red[tid + off];
        __syncthreads();
    }
    if (tid == 0) out[0] = red[0];
}

// ---------------------------------------------------------------------------
extern "C" void kernel_launch(void* const* d_in, const int* in_sizes, int n_in,
                              void* d_out, int out_size, void* d_ws, size_t ws_size,
                              hipStream_t stream) {
    const float* emissions = (const float*)d_in[0];
    const int*   tags      = (const int*)  d_in[1];
    const int*   mask      = (const int*)  d_in[2];
    const float* trans     = (const float*)d_in[3];
    const float* startT    = (const float*)d_in[4];
    const float* endT      = (const float*)d_in[5];
    float* out = (float*)d_out;

    float* ws_part  = (float*)d_ws;                 // [BATCH]
    float* ws_score = (float*)((char*)d_ws + 4096); // [BATCH]

    crf_score_kernel<<<BATCH / 256, 256, 0, stream>>>(
        emissions, tags, mask, trans, startT, endT, ws_score);

    // 64 workgroups x 1 wave; each wave owns 16 batch columns -> spread
    // the 64 independent recurrence chains across distinct WGPs/SIMDs.
    crf_partition_kernel<<<BATCH / 16, 32, 0, stream>>>(
        emissions, mask, trans, startT, endT, ws_part);

    crf_reduce_kernel<<<1, 256, 0, stream>>>(ws_part, ws_score, out);
}